// MambaMixer_89395449299144
// MI455X (gfx1250) — compile-verified
//
#include <hip/hip_runtime.h>
#include <math.h>

// ---------------------------------------------------------------------------
// Mamba mixer forward for MI455X (gfx1250), fp32 end-to-end.
// GEMMs use V_WMMA_F32_16X16X4_F32 (fp32 matrix core, wave32).
// ---------------------------------------------------------------------------

typedef float v2f __attribute__((ext_vector_type(2)));
typedef float v8f __attribute__((ext_vector_type(8)));

#define B_SZ    4
#define LSEQ    1024
#define DMODEL  1024
#define DINNER  2048
#define DSTATE  16
#define DCONV   4
#define DTRANK  64
#define NROWS   (B_SZ * LSEQ)   // 4096

__device__ __forceinline__ float silu_f(float x) { return x / (1.0f + __expf(-x)); }
__device__ __forceinline__ float softplus_f(float x) {
    return (x > 20.0f) ? x : log1pf(__expf(x));
}

__device__ __forceinline__ v8f wmma_f32(v2f a, v2f b, v8f c) {
    return __builtin_amdgcn_wmma_f32_16x16x4_f32(
               false, a, false, b, (short)0, c, false, false);
}

// ---------------------------------------------------------------------------
// Register-blocked fp32 WMMA GEMM: C(MxN) = A(MxK) * B(KxN), row-major.
// One wave computes a 32x64 macro-tile (2x4 grid of 16x16 WMMA tiles):
// per K-step: 2 A-fragment b64 loads + 8 B loads -> 8 WMMAs (high reuse).
// Block = 2 waves along N (block tile 32x128). M%32==0, N%128==0, K%4==0.
// EPI: 0 = plain store, 1 = softplus(c + bias[col])  (dt path)
// ---------------------------------------------------------------------------
template <int EPI>
__global__ __launch_bounds__(64) void gemm_wmma_blocked(
    const float* __restrict__ A, const float* __restrict__ B,
    float* __restrict__ C, const float* __restrict__ bias,
    int K, int lda, int ldb, int ldc)
{
    const int lane   = threadIdx.x & 31;
    const int wave   = threadIdx.x >> 5;
    const int half   = lane >> 4;             // selects K pair {0,1} vs {2,3}
    const int mr     = lane & 15;             // M row (A) / N col (B, C)
    const int tm     = blockIdx.y;                      // 32-row macro-tile
    const int tnBase = (blockIdx.x * 2 + wave) * 4;     // first of 4 N tiles

    const float* Arow0 = A + (size_t)(tm * 32 + mr) * lda;
    const float* Arow1 = Arow0 + (size_t)16 * lda;
    const float* Bc    = B + (size_t)(tnBase * 16 + mr);

    v8f c[2][4];
#pragma unroll
    for (int i = 0; i < 2; ++i)
#pragma unroll
        for (int j = 0; j < 4; ++j) c[i][j] = (v8f){};

#pragma unroll 4
    for (int k0 = 0; k0 < K; k0 += 4) {
        const int ka = k0 + 2 * half;
        v2f a0 = *(const v2f*)(Arow0 + ka);   // two consecutive K -> one b64
        v2f a1 = *(const v2f*)(Arow1 + ka);
        v2f b[4];
#pragma unroll
        for (int j = 0; j < 4; ++j) {
            b[j].x = Bc[(size_t)ka * ldb + j * 16];
            b[j].y = Bc[(size_t)(ka + 1) * ldb + j * 16];
        }
#pragma unroll
        for (int j = 0; j < 4; ++j) {
            c[0][j] = wmma_f32(a0, b[j], c[0][j]);
            c[1][j] = wmma_f32(a1, b[j], c[1][j]);
        }
    }

#pragma unroll
    for (int j = 0; j < 4; ++j) {
        const int col = (tnBase + j) * 16 + mr;
        const float bv = (EPI == 1) ? bias[col] : 0.0f;
#pragma unroll
        for (int i = 0; i < 2; ++i) {
#pragma unroll
            for (int r = 0; r < 8; ++r) {
                const int row = tm * 32 + i * 16 + half * 8 + r;
                float v = c[i][j][r];
                if (EPI == 1) v = softplus_f(v + bv);
                C[(size_t)row * ldc + col] = v;
            }
        }
    }
}

// ---------------------------------------------------------------------------
// Simple 1-tile-per-wave WMMA GEMM (for the N=96 x_proj GEMM).
// ---------------------------------------------------------------------------
__global__ __launch_bounds__(64) void gemm_wmma_simple(
    const float* __restrict__ A, const float* __restrict__ B,
    float* __restrict__ C, int K, int lda, int ldb, int ldc)
{
    const int lane = threadIdx.x & 31;
    const int wave = threadIdx.x >> 5;
    const int tn   = blockIdx.x * 2 + wave;
    const int tm   = blockIdx.y;
    const int half = lane >> 4;
    const int mr   = lane & 15;

    const float* Arow = A + (size_t)(tm * 16 + mr) * lda;
    const float* Bcol = B + (size_t)(tn * 16 + mr);

    v8f c = {};
#pragma unroll 8
    for (int k0 = 0; k0 < K; k0 += 4) {
        const int ka = k0 + 2 * half;
        v2f a = *(const v2f*)(Arow + ka);
        v2f b;
        b.x = Bcol[(size_t)ka * ldb];
        b.y = Bcol[(size_t)(ka + 1) * ldb];
        c = wmma_f32(a, b, c);
    }

    const int col = tn * 16 + mr;
#pragma unroll
    for (int r = 0; r < 8; ++r)
        C[(size_t)(tm * 16 + half * 8 + r) * ldc + col] = c[r];
}

// ---------------------------------------------------------------------------
// Causal depthwise conv1d (kernel 4) + SiLU.  x = xz[:, :, 0:2048]
// ---------------------------------------------------------------------------
__global__ __launch_bounds__(256) void conv_silu_kernel(
    const float* __restrict__ xz, const float* __restrict__ conv_w,
    const float* __restrict__ conv_b, float* __restrict__ xc)
{
    const int idx = blockIdx.x * 256 + threadIdx.x;   // over NROWS * DINNER
    const int d   = idx & (DINNER - 1);
    const int pos = idx >> 11;                        // row = b*L + l
    const int l   = pos & (LSEQ - 1);
    const int b   = pos >> 10;

    float acc = conv_b[d];
#pragma unroll
    for (int k = 0; k < DCONV; ++k) {
        const int ls = l - (DCONV - 1) + k;
        if (ls >= 0)
            acc += conv_w[d * DCONV + k] *
                   xz[((size_t)(b * LSEQ + ls)) * (2 * DINNER) + d];
    }
    xc[(size_t)pos * DINNER + d] = silu_f(acc);
}

// ---------------------------------------------------------------------------
// Selective scan + fused gating. One thread per (b, d) channel.
// h[16] and the A row live in registers. The 32 shared B_t/C_t floats have a
// wave-uniform address -> compiler scalarizes them to s_load broadcasts
// (no LDS, no barriers). u/dt/y accesses are stride-1 in d (coalesced).
// Epilogue fused: y = (scan + D*u) * silu(z).
// ---------------------------------------------------------------------------
__global__ __launch_bounds__(256) void scan_gate_kernel(
    const float* __restrict__ xc,   const float* __restrict__ dtm,
    const float* __restrict__ xdbl, const float* __restrict__ A_log,
    const float* __restrict__ xz,   const float* __restrict__ D_param,
    float* __restrict__ y)
{
    const int b = blockIdx.y;
    const int d = blockIdx.x * 256 + threadIdx.x;

    float Ar[DSTATE];
#pragma unroll
    for (int n = 0; n < DSTATE; ++n)
        Ar[n] = -__expf(A_log[d * DSTATE + n]);
    const float Dv = D_param[d];

    float h[DSTATE];
#pragma unroll
    for (int n = 0; n < DSTATE; ++n) h[n] = 0.0f;

    const size_t rowBase = (size_t)b * LSEQ;
    for (int t = 0; t < LSEQ; ++t) {
        const size_t row = rowBase + t;
        const float* bc = xdbl + row * (DTRANK + 2 * DSTATE) + DTRANK; // uniform

        const float u   = xc [row * DINNER + d];
        const float dtv = dtm[row * DINNER + d];
        // prefetch next timestep's operand rows (global_prefetch_b8)
        __builtin_prefetch(xc  + (row + 1) * DINNER + d, 0, 1);
        __builtin_prefetch(dtm + (row + 1) * DINNER + d, 0, 1);

        float acc = 0.0f;
#pragma unroll
        for (int n = 0; n < DSTATE; ++n) {
            const float dA = __expf(dtv * Ar[n]);          // v_exp_f32 (TRANS)
            h[n] = dA * h[n] + (dtv * bc[n]) * u;
            acc += bc[DSTATE + n] * h[n];
        }
        const float z = xz[row * (2 * DINNER) + DINNER + d];
        y[row * DINNER + d] = (acc + Dv * u) * silu_f(z);
    }
}

// ---------------------------------------------------------------------------
extern "C" void kernel_launch(void* const* d_in, const int* in_sizes, int n_in,
                              void* d_out, int out_size, void* d_ws, size_t ws_size,
                              hipStream_t stream)
{
    const float* hidden    = (const float*)d_in[0];   // (B,L,1024)
    const float* in_proj_w = (const float*)d_in[1];   // (1024,4096)
    const float* conv_w    = (const float*)d_in[2];   // (2048,4)
    const float* conv_b    = (const float*)d_in[3];   // (2048,)
    const float* x_proj_w  = (const float*)d_in[4];   // (2048,96)
    const float* dt_proj_w = (const float*)d_in[5];   // (64,2048)
    const float* dt_bias   = (const float*)d_in[6];   // (2048,)
    const float* A_log     = (const float*)d_in[7];   // (2048,16)
    const float* D_param   = (const float*)d_in[8];   // (2048,)
    const float* out_proj  = (const float*)d_in[9];   // (2048,1024)
    float*       out       = (float*)d_out;           // (B,L,1024)

    // Workspace layout (floats)
    float* ws   = (float*)d_ws;
    float* xz   = ws;                                   // 4096*4096
    float* xc   = xz   + (size_t)NROWS * 2 * DINNER;    // 4096*2048
    float* xdbl = xc   + (size_t)NROWS * DINNER;        // 4096*96
    float* dtm  = xdbl + (size_t)NROWS * (DTRANK + 2 * DSTATE); // 4096*2048
    float* y    = dtm  + (size_t)NROWS * DINNER;        // 4096*2048

    // 1) xz = hidden @ in_proj_w        (4096 x 4096, K=1024)
    gemm_wmma_blocked<0><<<dim3((2 * DINNER) / 128, NROWS / 32), 64, 0, stream>>>(
        hidden, in_proj_w, xz, nullptr, DMODEL, DMODEL, 2 * DINNER, 2 * DINNER);

    // 2) xc = silu(conv(x) + b)
    conv_silu_kernel<<<(NROWS * DINNER) / 256, 256, 0, stream>>>(xz, conv_w, conv_b, xc);

    // 3) xdbl = xc @ x_proj_w           (4096 x 96, K=2048)
    gemm_wmma_simple<<<dim3((DTRANK + 2 * DSTATE) / 32, NROWS / 16), 64, 0, stream>>>(
        xc, x_proj_w, xdbl, DINNER, DINNER,
        DTRANK + 2 * DSTATE, DTRANK + 2 * DSTATE);

    // 4) dtm = softplus(xdbl[:, :64] @ dt_proj_w + dt_bias)   (4096 x 2048, K=64)
    gemm_wmma_blocked<1><<<dim3(DINNER / 128, NROWS / 32), 64, 0, stream>>>(
        xdbl, dt_proj_w, dtm, dt_bias, DTRANK,
        DTRANK + 2 * DSTATE, DINNER, DINNER);

    // 5) selective scan + gating -> y
    scan_gate_kernel<<<dim3(DINNER / 256, B_SZ), 256, 0, stream>>>(
        xc, dtm, xdbl, A_log, xz, D_param, y);

    // 6) out = y @ out_proj_w           (4096 x 1024, K=2048)
    gemm_wmma_blocked<0><<<dim3(DMODEL / 128, NROWS / 32), 64, 0, stream>>>(
        y, out_proj, out, nullptr, DINNER, DINNER, DMODEL, DMODEL);
}